// StringGNNDEGModel_6923487282446
// MI455X (gfx1250) — compile-verified
//
#include <hip/hip_runtime.h>
#include <math.h>

#define N_GENES 6640
#define BATCH   32
#define DIM     256
#define HID     512
#define LN_EPS  1e-5f

typedef __attribute__((ext_vector_type(2))) float v2f;
typedef __attribute__((ext_vector_type(8))) float v8f;

__device__ __forceinline__ float wave_sum32(float v) {
#pragma unroll
    for (int off = 16; off; off >>= 1) v += __shfl_xor(v, off, 32);
    return v;
}

// ---------------------------------------------------------------------------
// Kernel 1: W1g[h,d] = w1[h,d]*gamma[d];  s[h] = sum_d W1g[h,d];
//           t[h] = sum_d beta[d]*w1[h,d] + b1[h]
// one wave per h; 8 waves/block, 64 blocks
// ---------------------------------------------------------------------------
__global__ void prep_w1_kernel(const float* __restrict__ w1,
                               const float* __restrict__ gamma,
                               const float* __restrict__ beta,
                               const float* __restrict__ b1,
                               float* __restrict__ W1g,
                               float* __restrict__ s,
                               float* __restrict__ t) {
    const int wave = threadIdx.x >> 5, lane = threadIdx.x & 31;
    const int h = blockIdx.x * 8 + wave;
    const float* row = w1 + (size_t)h * HID;
    float*      orow = W1g + (size_t)h * HID;
    float ss = 0.f, ts = 0.f;
#pragma unroll
    for (int j = 0; j < 16; ++j) {
        const int d = j * 32 + lane;
        const float w  = row[d];
        const float wg = w * gamma[d];
        orow[d] = wg;
        ss += wg;
        ts += w * beta[d];
    }
    ss = wave_sum32(ss);
    ts = wave_sum32(ts);
    if (lane == 0) { s[h] = ss; t[h] = ts + b1[h]; }
}

// ---------------------------------------------------------------------------
// Kernel 2: per-gene row sum & sum-of-squares.  one wave per gene.
// ---------------------------------------------------------------------------
__global__ void gene_stats_kernel(const float* __restrict__ gq,
                                  float* __restrict__ sq,
                                  float* __restrict__ ssq) {
    const int wave = threadIdx.x >> 5, lane = threadIdx.x & 31;
    const int g = blockIdx.x * 8 + wave;
    const float* row = gq + (size_t)g * DIM;
    float a = 0.f, a2 = 0.f;
#pragma unroll
    for (int j = 0; j < 8; ++j) {
        const float v = row[j * 32 + lane];
        a += v; a2 += v * v;
    }
    a  = wave_sum32(a);
    a2 = wave_sum32(a2);
    if (lane == 0) { sq[g] = a; ssq[g] = a2; }
}

// ---------------------------------------------------------------------------
// Kernel 3: gather pert rows + stats.  one wave per batch row, grid 4 x 256.
// ---------------------------------------------------------------------------
__global__ void pert_gather_kernel(const float* __restrict__ gnn,
                                   const int* __restrict__ pidx,
                                   float* __restrict__ pbuf,
                                   float* __restrict__ sp,
                                   float* __restrict__ ssp) {
    const int wave = threadIdx.x >> 5, lane = threadIdx.x & 31;
    const int b = blockIdx.x * 8 + wave;
    const float* row = gnn + (size_t)pidx[b] * DIM;
    float a = 0.f, a2 = 0.f;
#pragma unroll
    for (int j = 0; j < 8; ++j) {
        const int d = j * 32 + lane;
        const float v = row[d];
        pbuf[b * DIM + d] = v;
        a += v; a2 += v * v;
    }
    a  = wave_sum32(a);
    a2 = wave_sum32(a2);
    if (lane == 0) { sp[b] = a; ssp[b] = a2; }
}

// ---------------------------------------------------------------------------
// Kernel 4: Out[m,h] = sum_{k<256} A[m,k] * W1g[h, doff+k]
// fp32 WMMA 16x16x4, one 16x16 tile per wave, 4 waves/block.
// A: [Mtiles*16, 256] row-major.  Out: [Mtiles*16, 512] row-major.
// A frag (ISA 7.12.2, 32-bit A 16x4): lane m=lane&15, comps hold K =
// (lane>=16?2:0)+{0,1}.  D frag: vgpr v -> M = v + 8*(lane>>4), N = lane&15.
// ---------------------------------------------------------------------------
__global__ void gemm16_f32_wmma_kernel(const float* __restrict__ A,
                                       const float* __restrict__ W1g,
                                       int doff,
                                       float* __restrict__ Out) {
    const int wave = threadIdx.x >> 5, lane = threadIdx.x & 31;
    const int m0 = blockIdx.x * 16;
    const int h0 = (blockIdx.y * 4 + wave) * 16;
    const int r  = lane & 15;
    const int kb = (lane >> 4) << 1;                    // 0 or 2
    const float* arow = A   + (size_t)(m0 + r) * DIM + kb;
    const float* brow = W1g + (size_t)(h0 + r) * HID + doff + kb;

    v8f acc = {};
#pragma unroll 4
    for (int k = 0; k < DIM; k += 4) {
        v2f a = *(const v2f*)(arow + k);
        v2f b = *(const v2f*)(brow + k);
        acc = __builtin_amdgcn_wmma_f32_16x16x4_f32(
            false, a, false, b, (short)0, acc, false, false);
    }
    const int mrow = m0 + ((lane >> 4) << 3);
    const int col  = h0 + r;
#pragma unroll
    for (int v = 0; v < 8; ++v)
        Out[(size_t)(mrow + v) * HID + col] = acc[v];
}

// ---------------------------------------------------------------------------
// Kernel 5: epilogue.  one wave per (b,g) pair, 8 waves/block.
// z[h] = inv*(P[b,h] + Q[g,h] - mu*s[h]) + t[h]; gelu; dot with w2 rows.
// ---------------------------------------------------------------------------
__global__ void epilogue_kernel(const float* __restrict__ Q,
                                const float* __restrict__ P,
                                const float* __restrict__ s,
                                const float* __restrict__ t,
                                const float* __restrict__ sq,
                                const float* __restrict__ ssq,
                                const float* __restrict__ sp,
                                const float* __restrict__ ssp,
                                const float* __restrict__ w2,
                                const float* __restrict__ b2,
                                float* __restrict__ out) {
    __shared__ float sh_s[HID];
    __shared__ float sh_t[HID];
    __shared__ float sh_w2[3 * HID];
    for (int i = threadIdx.x; i < HID; i += blockDim.x) {
        sh_s[i] = s[i];
        sh_t[i] = t[i];
    }
    for (int i = threadIdx.x; i < 3 * HID; i += blockDim.x) sh_w2[i] = w2[i];
    __syncthreads();

    const int wave = threadIdx.x >> 5, lane = threadIdx.x & 31;
    const int pair = blockIdx.x * 8 + wave;             // 0..212479
    const int b = pair / N_GENES;
    const int g = pair % N_GENES;

    const float mu  = (sp[b] + sq[g]) * (1.f / (2.f * DIM));
    const float ms  = (ssp[b] + ssq[g]) * (1.f / (2.f * DIM));
    const float inv = rsqrtf(ms - mu * mu + LN_EPS);

    const float* qrow = Q + (size_t)g * HID;
    const float* prow = P + (size_t)b * HID;

    float a0 = 0.f, a1 = 0.f, a2 = 0.f;
#pragma unroll
    for (int i = 0; i < 16; ++i) {
        const int h = i * 32 + lane;
        const float z  = inv * (prow[h] + qrow[h] - mu * sh_s[h]) + sh_t[h];
        const float ge = 0.5f * z * (1.f + erff(z * 0.70710678118654752f));
        a0 += ge * sh_w2[h];
        a1 += ge * sh_w2[HID + h];
        a2 += ge * sh_w2[2 * HID + h];
    }
    a0 = wave_sum32(a0);
    a1 = wave_sum32(a1);
    a2 = wave_sum32(a2);
    if (lane == 0) {
        float* o = out + ((size_t)b * 3) * N_GENES + g;
        o[0]           = a0 + b2[0];
        o[N_GENES]     = a1 + b2[1];
        o[2 * N_GENES] = a2 + b2[2];
    }
}

// ---------------------------------------------------------------------------
// Workspace layout (float offsets)
// ---------------------------------------------------------------------------
#define WS_Q     0                                   // 6640*512 = 3399680
#define WS_W1G   (WS_Q + N_GENES * HID)              // 512*512  = 262144
#define WS_P     (WS_W1G + HID * HID)                // 32*512
#define WS_PERT  (WS_P + BATCH * HID)                // 32*256
#define WS_S     (WS_PERT + BATCH * DIM)             // 512
#define WS_T     (WS_S + HID)                        // 512
#define WS_SQ    (WS_T + HID)                        // 6640
#define WS_SSQ   (WS_SQ + N_GENES)                   // 6640
#define WS_SP    (WS_SSQ + N_GENES)                  // 32
#define WS_SSP   (WS_SP + BATCH)                     // 32

extern "C" void kernel_launch(void* const* d_in, const int* in_sizes, int n_in,
                              void* d_out, int out_size, void* d_ws, size_t ws_size,
                              hipStream_t stream) {
    const float* gnn   = (const float*)d_in[0];
    const float* gq    = (const float*)d_in[1];
    const float* gamma = (const float*)d_in[2];
    const float* beta  = (const float*)d_in[3];
    const float* w1    = (const float*)d_in[4];
    const float* b1    = (const float*)d_in[5];
    const float* w2    = (const float*)d_in[6];
    const float* b2    = (const float*)d_in[7];
    const int*   pidx  = (const int*)d_in[8];
    float* out = (float*)d_out;
    float* ws  = (float*)d_ws;

    float* Q    = ws + WS_Q;
    float* W1g  = ws + WS_W1G;
    float* P    = ws + WS_P;
    float* pert = ws + WS_PERT;
    float* s    = ws + WS_S;
    float* t    = ws + WS_T;
    float* sq   = ws + WS_SQ;
    float* ssq  = ws + WS_SSQ;
    float* sp   = ws + WS_SP;
    float* ssp  = ws + WS_SSP;

    // 1) W1g = gamma (x) w1, plus s/t reductions   (512 waves)
    prep_w1_kernel<<<64, 256, 0, stream>>>(w1, gamma, beta, b1, W1g, s, t);
    // 2) per-gene stats                             (6640 waves)
    gene_stats_kernel<<<N_GENES / 8, 256, 0, stream>>>(gq, sq, ssq);
    // 3) gather pert rows + stats                   (32 waves)
    pert_gather_kernel<<<BATCH / 8, 256, 0, stream>>>(gnn, pidx, pert, sp, ssp);
    // 4) Q = gene_query @ W1g_hi^T   (415 x 32 tiles, fp32 WMMA)
    gemm16_f32_wmma_kernel<<<dim3(N_GENES / 16, 8), 128, 0, stream>>>(gq, W1g, DIM, Q);
    //    P = pert @ W1g_lo^T          (2 x 32 tiles)
    gemm16_f32_wmma_kernel<<<dim3(BATCH / 16, 8), 128, 0, stream>>>(pert, W1g, 0, P);
    // 5) fused LN-correction + GELU + 512->3        (212480 waves)
    epilogue_kernel<<<(BATCH * N_GENES) / 8, 256, 0, stream>>>(
        Q, P, s, t, sq, ssq, sp, ssp, w2, b2, out);
}